// Network_70918499991665
// MI455X (gfx1250) — compile-verified
//
#include <hip/hip_runtime.h>
#include <hip/hip_bf16.h>
#include <math.h>

#define BETA 100.0f
#define NPTS 4096
#define N4   16384   // 4 * NPTS columns: per point {x, gx, gy, gz}

#define KC    16     // K-chunk staged in LDS per iteration
#define LSTR  20     // LDS row stride in floats: conflict-free & 16B-aligned stores

typedef float v2f __attribute__((ext_vector_type(2)));
typedef float v8f __attribute__((ext_vector_type(8)));

// ---------------------------------------------------------------------------
// activation helpers (stable softplus with beta=100)
// ---------------------------------------------------------------------------
__device__ __forceinline__ float softplus_b(float pre) {
    float z = BETA * pre;
    if (z > 0.0f) return pre + log1pf(__expf(-z)) * (1.0f / BETA);
    return log1pf(__expf(z)) * (1.0f / BETA);
}
__device__ __forceinline__ float sigmoid_b(float pre) {
    return 1.0f / (1.0f + __expf(-BETA * pre));
}

// ---------------------------------------------------------------------------
// K_con: write input_con = concat(input, broadcast latent) to d_out segment
// ---------------------------------------------------------------------------
__global__ __launch_bounds__(256) void k_write_con(const float* __restrict__ input,
                                                   const float* __restrict__ latent,
                                                   float* __restrict__ dst) {
    int idx = blockIdx.x * 256 + threadIdx.x;
    if (idx >= NPTS * 259) return;
    int n = idx / 259, i = idx % 259;
    dst[idx] = (i < 3) ? input[n * 3 + i] : latent[i - 3];
}

// ---------------------------------------------------------------------------
// K_c1prep: weight-norm first conv (515x259); fold latent dot + bias into a[c]
// (B=1 so the latent contribution is a per-channel constant). one wave per row.
// ---------------------------------------------------------------------------
__global__ __launch_bounds__(256) void k_c1_prep(const float* __restrict__ v,
                                                 const float* __restrict__ g,
                                                 const float* __restrict__ b,
                                                 const float* __restrict__ latent,
                                                 float* __restrict__ wxyz,
                                                 float* __restrict__ avec) {
    int wave = threadIdx.x >> 5, lane = threadIdx.x & 31;
    int c = blockIdx.x * 8 + wave;
    if (c >= 515) return;
    const float* row = v + c * 259;
    float ss = 0.0f;
    for (int k = lane; k < 259; k += 32) { float t = row[k]; ss += t * t; }
    for (int o = 16; o; o >>= 1) ss += __shfl_down(ss, o, 32);
    ss = __shfl(ss, 0, 32);
    float scale = g[c] / sqrtf(ss);
    float dot = 0.0f;
    for (int k = lane; k < 256; k += 32) dot += row[3 + k] * scale * latent[k];
    for (int o = 16; o; o >>= 1) dot += __shfl_down(dot, o, 32);
    if (lane < 3) wxyz[c * 3 + lane] = row[lane] * scale;
    if (lane == 0) avec[c] = dot + b[c];
}

// ---------------------------------------------------------------------------
// K_first: first layer expand. S0 column-major [N4][KP0], col = 4n+d.
// ---------------------------------------------------------------------------
__global__ __launch_bounds__(256) void k_first(const float* __restrict__ pts,
                                               const float* __restrict__ wxyz,
                                               const float* __restrict__ avec,
                                               float* __restrict__ S0, int KP) {
    int idx = blockIdx.x * 256 + threadIdx.x;
    int c = idx % KP, n = idx / KP;
    if (n >= NPTS) return;
    size_t base = (size_t)(4 * n) * KP + c;
    if (c >= 515) {
        S0[base] = 0.0f; S0[base + KP] = 0.0f;
        S0[base + 2 * (size_t)KP] = 0.0f; S0[base + 3 * (size_t)KP] = 0.0f;
        return;
    }
    float wx = wxyz[c * 3 + 0], wy = wxyz[c * 3 + 1], wz = wxyz[c * 3 + 2];
    float pre = avec[c] + wx * pts[n * 3 + 0] + wy * pts[n * 3 + 1] + wz * pts[n * 3 + 2];
    float sg = sigmoid_b(pre);
    S0[base]                  = softplus_b(pre);
    S0[base + KP]             = sg * wx;
    S0[base + 2 * (size_t)KP] = sg * wy;
    S0[base + 3 * (size_t)KP] = sg * wz;
}

// ---------------------------------------------------------------------------
// K_wnorm: weight-norm into padded row-major Wn[M][KP]. one wave per row.
// ---------------------------------------------------------------------------
__global__ __launch_bounds__(256) void k_wnorm(const float* __restrict__ v,
                                               const float* __restrict__ g,
                                               float* __restrict__ Wn,
                                               int M, int K, int KP) {
    int wave = threadIdx.x >> 5, lane = threadIdx.x & 31;
    int m = blockIdx.x * 8 + wave;
    if (m >= M) return;
    const float* row = v + (size_t)m * K;
    float ss = 0.0f;
    for (int k = lane; k < K; k += 32) { float t = row[k]; ss += t * t; }
    for (int o = 16; o; o >>= 1) ss += __shfl_down(ss, o, 32);
    ss = __shfl(ss, 0, 32);
    float scale = g[m] / sqrtf(ss);
    float* orow = Wn + (size_t)m * KP;
    for (int k = lane; k < KP; k += 32) orow[k] = (k < K) ? row[k] * scale : 0.0f;
}

// ---------------------------------------------------------------------------
// K_gemm: LDS-tiled, double-buffered fused GEMM + activation epilogue.
//   Block = 8 waves = 128(M) x 128(col) macro-tile; wave = 16(M) x 128(col).
//   Per K-chunk of 16: stage A[128x16], B[128x16] panels into LDS (coalesced
//   float4, zero-filled beyond M), one barrier per chunk, register prefetch
//   of the next chunk issued before the WMMA phase.
//   Sin col-major [N4][KP], Wn row-major [M][KP], Sout col-major [N4][M].
// ---------------------------------------------------------------------------
__global__ __launch_bounds__(256) void k_gemm_act(const float* __restrict__ Wn,
                                                  const float* __restrict__ bias,
                                                  const float* __restrict__ Sin,
                                                  float* __restrict__ Sout,
                                                  int M, int KP) {
    __shared__ float Ash[2][128 * LSTR];
    __shared__ float Bsh[2][128 * LSTR];

    const int tid  = threadIdx.x;
    const int wave = tid >> 5;
    const int lane = tid & 31;
    const int n0   = blockIdx.x * 128;                 // 8 tiles of 16 columns
    const int mB   = blockIdx.y * 128;
    const int m0   = mB + wave * 16;
    const bool active = (m0 < M);                      // uniform per wave

    const int row = lane & 15;                         // A row / B,C column idx
    const int hi  = lane >> 4;                         // upper half-wave flag
    const int d   = lane & 3;                          // component within point

    // cooperative staging: 128 rows/cols x 16 k, 2 float4 per thread each
    const int crow = tid >> 1;                         // 0..127
    const int chv  = (tid & 1) * 8;                    // 0 or 8 (floats)
    const float* gB = Sin + (size_t)(n0 + crow) * KP + chv;
    const int agr   = mB + crow;
    const float* gA = Wn + (size_t)agr * KP + chv;
    const bool aok  = (agr < M);

    const int nch = KP / KC;

    float4 rB0, rB1, rA0, rA1;
    const float4 z4 = make_float4(0.f, 0.f, 0.f, 0.f);

    // prologue: chunk 0 -> buffer 0
    rB0 = *(const float4*)(gB);
    rB1 = *(const float4*)(gB + 4);
    rA0 = aok ? *(const float4*)(gA)     : z4;
    rA1 = aok ? *(const float4*)(gA + 4) : z4;
    *(float4*)&Bsh[0][crow * LSTR + chv]     = rB0;
    *(float4*)&Bsh[0][crow * LSTR + chv + 4] = rB1;
    *(float4*)&Ash[0][crow * LSTR + chv]     = rA0;
    *(float4*)&Ash[0][crow * LSTR + chv + 4] = rA1;
    __syncthreads();

    v8f acc[8];
    #pragma unroll
    for (int t = 0; t < 8; ++t) acc[t] = (v8f){0.f,0.f,0.f,0.f,0.f,0.f,0.f,0.f};

    const int aoff = (wave * 16 + row) * LSTR + hi * 2;
    const int boff = row * LSTR + hi * 2;

    for (int c = 0; c < nch; ++c) {
        const int nb = c & 1;
        const bool more = (c + 1 < nch);
        if (more) {                                    // prefetch next chunk
            const int kc = (c + 1) * KC;
            rB0 = *(const float4*)(gB + kc);
            rB1 = *(const float4*)(gB + kc + 4);
            rA0 = aok ? *(const float4*)(gA + kc)     : z4;
            rA1 = aok ? *(const float4*)(gA + kc + 4) : z4;
        }
        if (active) {
            const float* As = &Ash[nb][aoff];
            const float* Bs = &Bsh[nb][boff];
            #pragma unroll
            for (int kl = 0; kl < KC; kl += 4) {
                v2f a = *(const v2f*)(As + kl);
                #pragma unroll
                for (int t = 0; t < 8; ++t) {
                    v2f bfrag = *(const v2f*)(Bs + t * 16 * LSTR + kl);
                    acc[t] = __builtin_amdgcn_wmma_f32_16x16x4_f32(
                        false, a, false, bfrag, (short)0, acc[t], false, false);
                }
            }
        }
        if (more) {                                    // stage next chunk
            const int nb2 = nb ^ 1;
            *(float4*)&Bsh[nb2][crow * LSTR + chv]     = rB0;
            *(float4*)&Bsh[nb2][crow * LSTR + chv + 4] = rB1;
            *(float4*)&Ash[nb2][crow * LSTR + chv]     = rA0;
            *(float4*)&Ash[nb2][crow * LSTR + chv + 4] = rA1;
        }
        __syncthreads();
    }
    if (!active) return;

    // epilogue: bias + softplus on d==0 column, sigmoid(beta*pre_d0)*raw on grads
    float bv[8];
    #pragma unroll
    for (int vr = 0; vr < 8; ++vr) bv[vr] = bias[m0 + hi * 8 + vr];

    const int srcLane = lane & ~3;                     // d==0 lane, same row half
    #pragma unroll
    for (int t = 0; t < 8; ++t) {
        const int col = n0 + t * 16 + row;
        float out[8];
        #pragma unroll
        for (int vr = 0; vr < 8; ++vr) {
            float raw  = acc[t][vr];
            float pre  = raw + bv[vr];
            float pre0 = __shfl(pre, srcLane, 32);     // pre of d==0 col, same row
            float sg   = sigmoid_b(pre0);
            out[vr] = (d == 0) ? softplus_b(pre) : sg * raw;
        }
        float* dst = Sout + (size_t)col * M + m0 + hi * 8;
        *(float4*)(dst)     = make_float4(out[0], out[1], out[2], out[3]);
        *(float4*)(dst + 4) = make_float4(out[4], out[5], out[6], out[7]);
    }
}

// ---------------------------------------------------------------------------
// K_c10prep: weight-norm the 1x896 head (single wave)
// ---------------------------------------------------------------------------
__global__ __launch_bounds__(32) void k_c10_prep(const float* __restrict__ v,
                                                 const float* __restrict__ g,
                                                 float* __restrict__ w10n) {
    int lane = threadIdx.x;
    float ss = 0.0f;
    for (int k = lane; k < 896; k += 32) { float t = v[k]; ss += t * t; }
    for (int o = 16; o; o >>= 1) ss += __shfl_down(ss, o, 32);
    ss = __shfl(ss, 0, 32);
    float scale = g[0] / sqrtf(ss);
    for (int k = lane; k < 896; k += 32) w10n[k] = v[k] * scale;
}

// ---------------------------------------------------------------------------
// K_head: final 896 -> 1 projection for all 16384 columns. one wave per column.
// ---------------------------------------------------------------------------
__global__ __launch_bounds__(256) void k_head(const float* __restrict__ S,
                                              const float* __restrict__ w10n,
                                              const float* __restrict__ b10,
                                              float* __restrict__ out,
                                              float* __restrict__ outgrad) {
    int wave = threadIdx.x >> 5, lane = threadIdx.x & 31;
    int col = blockIdx.x * 8 + wave;
    if (col >= N4) return;
    const float* s = S + (size_t)col * 896;
    float dot = 0.0f;
    for (int k = lane; k < 896; k += 32) dot += w10n[k] * s[k];
    for (int o = 16; o; o >>= 1) dot += __shfl_down(dot, o, 32);
    if (lane == 0) {
        int n = col >> 2, dd = col & 3;
        if (dd == 0) out[n] = dot + b10[0];
        else         outgrad[n * 3 + dd - 1] = dot;
    }
}

// ---------------------------------------------------------------------------
// host launcher
// ---------------------------------------------------------------------------
struct LayerDesc { int vi, gi, bi, Cin, Cout, KPin; };

extern "C" void kernel_launch(void* const* d_in, const int* in_sizes, int n_in,
                              void* d_out, int out_size, void* d_ws, size_t ws_size,
                              hipStream_t stream) {
    (void)in_sizes; (void)n_in; (void)out_size; (void)ws_size;

    const float* input  = (const float*)d_in[0];
    const float* latent = (const float*)d_in[1];
    const float* c1_v   = (const float*)d_in[2];
    const float* c1_g   = (const float*)d_in[3];
    const float* c1_b   = (const float*)d_in[4];
    const float* c10_v  = (const float*)d_in[35];
    const float* c10_g  = (const float*)d_in[36];
    const float* c10_b  = (const float*)d_in[37];

    static const LayerDesc layers[10] = {
        { 5,  6,  7, 515, 512, 528},
        { 8,  9, 10, 512, 512, 512},
        {11, 12, 13, 512, 576, 512},
        {14, 15, 16, 576, 576, 576},
        {17, 18, 19, 576, 768, 576},
        {20, 21, 22, 768, 768, 768},
        {23, 24, 25, 768, 768, 768},
        {26, 27, 28, 768, 960, 768},
        {29, 30, 31, 960, 960, 960},
        {32, 33, 34, 960, 896, 960},
    };

    // workspace layout (floats)
    const size_t S_ELEMS = (size_t)N4 * 960;
    float* ws    = (float*)d_ws;
    float* Sbuf[2] = { ws, ws + S_ELEMS };
    float* Wn    = ws + 2 * S_ELEMS;
    float* wxyz  = Wn + (size_t)960 * 960;
    float* avec  = wxyz + 515 * 3;
    float* w10n  = avec + 515;

    float* out_val  = (float*)d_out;               // [4096]
    float* out_grad = out_val + NPTS;              // [4096*3]
    float* out_con  = out_grad + NPTS * 3;         // [4096*259]

    // 1) input_con output (independent)
    k_write_con<<<(NPTS * 259 + 255) / 256, 256, 0, stream>>>(input, latent, out_con);

    // 2) first-layer prep + expand into Sbuf[0] (KP0 = 528, channels 515..527 zero)
    k_c1_prep<<<(515 + 7) / 8, 256, 0, stream>>>(c1_v, c1_g, c1_b, latent, wxyz, avec);
    k_first<<<(NPTS * 528 + 255) / 256, 256, 0, stream>>>(input, wxyz, avec, Sbuf[0], 528);

    // 3) 10 fused resnext layers, ping-pong
    int cur = 0;
    for (int li = 0; li < 10; ++li) {
        const LayerDesc& L = layers[li];
        const float* v = (const float*)d_in[L.vi];
        const float* g = (const float*)d_in[L.gi];
        const float* b = (const float*)d_in[L.bi];
        k_wnorm<<<(L.Cout + 7) / 8, 256, 0, stream>>>(v, g, Wn, L.Cout, L.Cin, L.KPin);
        dim3 grid(N4 / 128, (L.Cout + 127) / 128);
        k_gemm_act<<<grid, 256, 0, stream>>>(Wn, b, Sbuf[cur], Sbuf[1 - cur], L.Cout, L.KPin);
        cur = 1 - cur;
    }

    // 4) head: 896 -> 1 (value + grad), no activation
    k_c10_prep<<<1, 32, 0, stream>>>(c10_v, c10_g, w10n);
    k_head<<<N4 / 8, 256, 0, stream>>>(Sbuf[cur], w10n, c10_b, out_val, out_grad);
}